// SurrogateGCN_39986145525889
// MI455X (gfx1250) — compile-verified
//
#include <hip/hip_runtime.h>
#include <math.h>

#define NODES_EXPECT 100000
#define IN_CH 128
#define HID_CH 128

typedef __attribute__((ext_vector_type(2))) float v2f;
typedef __attribute__((ext_vector_type(8))) float v8f;

// ---------------- degree / normalization ----------------

__global__ void gcn_deg_init(float* __restrict__ deg, int n) {
    int i = blockIdx.x * blockDim.x + threadIdx.x;
    if (i < n) deg[i] = 1.0f;  // self-loop contributes 1
}

__global__ void gcn_deg_count(const int* __restrict__ dst, float* __restrict__ deg, int e) {
    int i = blockIdx.x * blockDim.x + threadIdx.x;
    if (i < e) atomicAdd(&deg[dst[i]], 1.0f);
}

__global__ void gcn_dinv(float* __restrict__ deg, int n) {
    int i = blockIdx.x * blockDim.x + threadIdx.x;
    if (i < n) deg[i] = rsqrtf(deg[i]);  // deg >= 1 always (self-loops)
}

// ---------------- GEMM: C[M,128] = A[M,128] @ W[128,128] via V_WMMA_F32_16X16X4_F32 ----
// One wave -> one 16x16 output tile. Block = 8 waves = 16 rows x 128 cols.
// A-operand layout (ISA 7.12.2, 32-bit A 16x4): lanes 0-15 hold K=0/1, lanes 16-31 K=2/3.
// B-operand mirrors with N replacing M. C/D: v8f, row = i + 8*(lane>>4), col = lane&15.

__global__ __launch_bounds__(256) void gcn_gemm_wmma(const float* __restrict__ A,
                                                     const float* __restrict__ W,
                                                     float* __restrict__ C) {
    const int m0   = blockIdx.x * 16;
    const int wave = threadIdx.x >> 5;    // 0..7 -> N tile
    const int lane = threadIdx.x & 31;
    const int half = lane >> 4;           // 0 or 1
    const int r    = lane & 15;
    const int n0   = wave * 16;

    const float* __restrict__ arow = A + (size_t)(m0 + r) * IN_CH;
    v8f acc = {};
#pragma unroll
    for (int k = 0; k < IN_CH; k += 4) {
        const int ka = k + 2 * half;
        v2f a, b;
        a.x = arow[ka];
        a.y = arow[ka + 1];
        b.x = W[(size_t)ka * HID_CH + n0 + r];
        b.y = W[(size_t)(ka + 1) * HID_CH + n0 + r];
        acc = __builtin_amdgcn_wmma_f32_16x16x4_f32(
            /*neg_a=*/false, a, /*neg_b=*/false, b,
            /*c_mod=*/(short)0, acc, /*reuse_a=*/false, /*reuse_b=*/false);
    }
#pragma unroll
    for (int i = 0; i < 8; ++i) {
        C[(size_t)(m0 + i + 8 * half) * HID_CH + n0 + r] = acc[i];
    }
}

// ---------------- aggregation ----------------
// out[i] = h[i] * dinv[i]^2   (self-loop term, also initializes the buffer)
__global__ void gcn_agg_init(const float4* __restrict__ h, const float* __restrict__ dinv,
                             float4* __restrict__ out, int n) {
    int t = blockIdx.x * blockDim.x + threadIdx.x;  // n*32 threads, float4 each
    if (t >= n * 32) return;
    int node = t >> 5;
    float s = dinv[node];
    s = s * s;
    float4 v = h[t];
    v.x *= s; v.y *= s; v.z *= s; v.w *= s;
    out[t] = v;
}

// out[dst] += h[src] * dinv[src]*dinv[dst] ; one wave per edge, float4 per lane
__global__ void gcn_agg_edges(const float* __restrict__ h, const float* __restrict__ dinv,
                              const int* __restrict__ src, const int* __restrict__ dst,
                              float* __restrict__ out, int e) {
    int t = blockIdx.x * blockDim.x + threadIdx.x;
    if (t >= e * 32) return;
    int edge = t >> 5;
    int lane = t & 31;
    int s = src[edge];
    int d = dst[edge];
    float nrm = dinv[s] * dinv[d];
    float4 v = *((const float4*)(h + (size_t)s * HID_CH) + lane);
    float* op = out + (size_t)d * HID_CH + lane * 4;
    atomicAdd(op + 0, v.x * nrm);
    atomicAdd(op + 1, v.y * nrm);
    atomicAdd(op + 2, v.z * nrm);
    atomicAdd(op + 3, v.w * nrm);
}

// ---------------- epilogues ----------------
// hres = relu(agg1 + b1) + relu(hfc + be)   (hres may alias hfc; same-index in-place)
__global__ void gcn_combine1(const float* __restrict__ agg1, const float* __restrict__ b1,
                             const float* __restrict__ hfc, const float* __restrict__ be,
                             float* __restrict__ hres, int total) {
    int t = blockIdx.x * blockDim.x + threadIdx.x;
    if (t >= total) return;
    int c = t & (HID_CH - 1);
    float a = fmaxf(agg1[t] + b1[c], 0.0f);
    float f = fmaxf(hfc[t] + be[c], 0.0f);
    hres[t] = a + f;
}

__global__ void gcn_zero_pooled(float* __restrict__ pooled) {
    pooled[threadIdx.x] = 0.0f;
}

// sum over nodes of ( relu(agg2 + b2) + hres ), block covers 256 nodes, thread = channel
__global__ __launch_bounds__(128) void gcn_combine2_pool(const float* __restrict__ agg2,
                                                         const float* __restrict__ b2,
                                                         const float* __restrict__ hres,
                                                         float* __restrict__ pooled, int n) {
    int c = threadIdx.x;             // 0..127
    int n0 = blockIdx.x * 256;
    int n1 = min(n0 + 256, n);
    float bc = b2[c];
    float s = 0.0f;
    for (int node = n0; node < n1; ++node) {
        size_t idx = (size_t)node * HID_CH + c;
        s += fmaxf(agg2[idx] + bc, 0.0f) + hres[idx];
    }
    atomicAdd(&pooled[c], s);
}

__global__ void gcn_final(const float* __restrict__ pooled, const float* __restrict__ Wfc,
                          const float* __restrict__ bfc, float* __restrict__ out, int n) {
    if (threadIdx.x == 0 && blockIdx.x == 0) {
        float acc = bfc[0];
        float invn = 1.0f / (float)n;
        for (int c = 0; c < HID_CH; ++c) acc += (pooled[c] * invn) * Wfc[c];
        out[0] = 1.0f / (1.0f + __expf(-acc));
    }
}

// ---------------- launch ----------------

extern "C" void kernel_launch(void* const* d_in, const int* in_sizes, int n_in,
                              void* d_out, int out_size, void* d_ws, size_t ws_size,
                              hipStream_t stream) {
    const float* x   = (const float*)d_in[0];
    const int*   ei  = (const int*)d_in[1];
    const float* W1  = (const float*)d_in[2];
    const float* b1  = (const float*)d_in[3];
    const float* W2  = (const float*)d_in[4];
    const float* b2  = (const float*)d_in[5];
    const float* We  = (const float*)d_in[6];
    const float* be  = (const float*)d_in[7];
    const float* Wfc = (const float*)d_in[8];
    const float* bfc = (const float*)d_in[9];

    const int n = in_sizes[0] / IN_CH;   // 100000
    const int e = in_sizes[1] / 2;       // 1600000
    const int* src = ei;
    const int* dst = ei + e;

    // workspace carve-out (256B aligned slabs)
    char* ws = (char*)d_ws;
    size_t off = 0;
    auto carve = [&](size_t bytes) -> void* {
        void* p = ws + off;
        off += (bytes + 255) & ~(size_t)255;
        return p;
    };
    float* dinv   = (float*)carve((size_t)n * sizeof(float));
    float* bufA   = (float*)carve((size_t)n * HID_CH * sizeof(float)); // gemm out
    float* bufB   = (float*)carve((size_t)n * HID_CH * sizeof(float)); // aggregation
    float* bufC   = (float*)carve((size_t)n * HID_CH * sizeof(float)); // x_fc -> hres
    float* pooled = (float*)carve(HID_CH * sizeof(float));
    (void)ws_size;

    const int B = 256;
    const int nB    = (n + B - 1) / B;
    const int eB    = (e + B - 1) / B;
    const int n32B  = (n * 32 + B - 1) / B;
    const int e32B  = (e * 32 + B - 1) / B;
    const int nhB   = (n * HID_CH + B - 1) / B;
    const int mTile = n / 16;  // 6250, exact

    // degree + dinv (in place)
    gcn_deg_init<<<nB, B, 0, stream>>>(dinv, n);
    gcn_deg_count<<<eB, B, 0, stream>>>(dst, dinv, e);
    gcn_dinv<<<nB, B, 0, stream>>>(dinv, n);

    // layer 1: H1 = x @ W1 ; agg -> bufB
    gcn_gemm_wmma<<<mTile, B, 0, stream>>>(x, W1, bufA);
    gcn_agg_init<<<n32B, B, 0, stream>>>((const float4*)bufA, dinv, (float4*)bufB, n);
    gcn_agg_edges<<<e32B, B, 0, stream>>>(bufA, dinv, src, dst, bufB, e);

    // x_fc path: Hfc = x @ We -> bufC ; hres = relu(agg1+b1) + relu(Hfc+be) -> bufC
    gcn_gemm_wmma<<<mTile, B, 0, stream>>>(x, We, bufC);
    gcn_combine1<<<nhB, B, 0, stream>>>(bufB, b1, bufC, be, bufC, n * HID_CH);

    // layer 2: H2 = hres @ W2 -> bufA ; agg -> bufB
    gcn_gemm_wmma<<<mTile, B, 0, stream>>>(bufC, W2, bufA);
    gcn_agg_init<<<n32B, B, 0, stream>>>((const float4*)bufA, dinv, (float4*)bufB, n);
    gcn_agg_edges<<<e32B, B, 0, stream>>>(bufA, dinv, src, dst, bufB, e);

    // mean pool + fc + sigmoid
    gcn_zero_pooled<<<1, HID_CH, 0, stream>>>(pooled);
    gcn_combine2_pool<<<(n + 255) / 256, HID_CH, 0, stream>>>(bufB, b2, bufC, pooled, n);
    gcn_final<<<1, 32, 0, stream>>>(pooled, Wfc, bfc, (float*)d_out, n);
}